// StatsBaseline_75376676045074
// MI455X (gfx1250) — compile-verified
//
#include <hip/hip_runtime.h>

typedef float v2f __attribute__((ext_vector_type(2)));
typedef float v8f __attribute__((ext_vector_type(8)));

#define S_LEN 4096
#define TOK 64
#define EFFD 60
#define CHUNKS 16
#define ROWS_PER_CHUNK (S_LEN / CHUNKS)   // 256

// ---------------------------------------------------------------------------
// Kernel 1: bandwidth-bound partial mean over the sequence axis.
// Each block handles (batch b, chunk of 256 rows). float4 (b128) loads,
// fully coalesced; deterministic serial LDS reduction; partials -> d_ws.
// ---------------------------------------------------------------------------
__global__ __launch_bounds__(256) void mean_partial_kernel(
    const float* __restrict__ x, float* __restrict__ part) {
  const int b      = blockIdx.x;
  const int chunk  = blockIdx.y;
  const int lane16 = threadIdx.x & 15;   // channel quad: 4*lane16 .. +3
  const int sub    = threadIdx.x >> 4;   // 0..15 row subgroup

  const float4* __restrict__ xr =
      (const float4*)(x + ((size_t)b * S_LEN + (size_t)chunk * ROWS_PER_CHUNK) * TOK);

  float4 acc = {0.f, 0.f, 0.f, 0.f};
  for (int i = sub; i < ROWS_PER_CHUNK; i += 16) {
    float4 v = xr[i * (TOK / 4) + lane16];
    acc.x += v.x; acc.y += v.y; acc.z += v.z; acc.w += v.w;
  }

  __shared__ float4 red[16][16];
  red[sub][lane16] = acc;
  __syncthreads();
  if (sub == 0) {
    float4 t = red[0][lane16];
#pragma unroll
    for (int j = 1; j < 16; ++j) {
      float4 v = red[j][lane16];
      t.x += v.x; t.y += v.y; t.z += v.z; t.w += v.w;
    }
    ((float4*)part)[((size_t)b * CHUNKS + chunk) * (TOK / 4) + lane16] = t;
  }
}

// ---------------------------------------------------------------------------
// WMMA f32 16x16x4 layer helper.
//   in:   LDS activations, row-major [16 x ldIn], K = Kdim valid cols
//   w:    global weights, row-major [Ndim x Kdim]  (B[k][n] = w[n*Kdim+k])
//   out:  LDS, row-major [16 x ldOut]; writes cols < Ndim only
// A-frag (16x4 fp32): lane r=lane&15 is row M; v0 = A[r][k0+2*(lane>>4)],
// v1 = A[r][k0+2*(lane>>4)+1].  B-frag (4x16): symmetric with N = lane&15.
// No divergent branches around the WMMA (EXEC stays all-ones).
// ---------------------------------------------------------------------------
__device__ __forceinline__ void wmma_layer(
    const float* __restrict__ in, int ldIn, int Kdim,
    const float* __restrict__ w, const float* __restrict__ bias, int Ndim,
    float* __restrict__ outLds, int ldOut, bool relu,
    int lane, int wave, int nWaves) {
  const int ntiles = (Ndim + 15) >> 4;
  const int r      = lane & 15;          // A row / B col within tile
  const int khalf  = (lane >> 4) << 1;   // 0 or 2

  for (int nt = wave; nt < ntiles; nt += nWaves) {
    const int   n      = nt * 16 + r;
    const bool  nvalid = (n < Ndim);
    const int   nc     = nvalid ? n : 0;     // clamped, select not branch
    const float nmask  = nvalid ? 1.0f : 0.0f;
    const float* __restrict__ wrow = w + (size_t)nc * Kdim;

    v8f acc = {};
    for (int k0 = 0; k0 < Kdim; k0 += 4) {
      v2f a, bb;
      a.x  = in[r * ldIn + k0 + khalf];
      a.y  = in[r * ldIn + k0 + khalf + 1];
      bb.x = wrow[k0 + khalf] * nmask;
      bb.y = wrow[k0 + khalf + 1] * nmask;
      acc = __builtin_amdgcn_wmma_f32_16x16x4_f32(
          /*neg_a=*/false, a, /*neg_b=*/false, bb,
          /*c_mod=*/(short)0, acc, /*reuse_a=*/false, /*reuse_b=*/false);
    }

    const float bv = bias[nc] * nmask;
#pragma unroll
    for (int g = 0; g < 8; ++g) {
      const int row = g + ((lane >> 4) << 3);   // C/D: VGPR g -> M=g / M=g+8
      float v = acc[g] + bv;
      if (relu) v = fmaxf(v, 0.0f);
      if (nvalid) outLds[row * ldOut + n] = v;
    }
  }
}

// ---------------------------------------------------------------------------
// Kernel 2: reduce chunk partials -> xm tile, then full MLP via WMMA.
// One block per 16-row M-tile (8 blocks), 4 waves per block.
// LDS ping-pong: actA [16x256], actB [16x128].
// ---------------------------------------------------------------------------
__global__ __launch_bounds__(128) void mlp_wmma_kernel(
    const float* __restrict__ part,
    const float* __restrict__ w_ext, const float* __restrict__ b_ext,
    const float* __restrict__ w0, const float* __restrict__ b0,
    const float* __restrict__ w1, const float* __restrict__ b1,
    const float* __restrict__ w2, const float* __restrict__ b2,
    const float* __restrict__ w3, const float* __restrict__ b3,
    float* __restrict__ out) {
  __shared__ float actA[16 * 256];
  __shared__ float actB[16 * 128];

  const int tid  = threadIdx.x;
  const int lane = tid & 31;
  const int wave = tid >> 5;   // 0..3
  const int mt   = blockIdx.x; // rows mt*16 .. mt*16+15

  // Step 0: sum the 16 chunk partials -> xm, scale by 1/S. actA ld=64.
  for (int idx = tid; idx < 16 * TOK; idx += 128) {
    const int r = idx >> 6;
    const int c = idx & 63;
    const int b = mt * 16 + r;
    float s = 0.f;
    for (int cc = 0; cc < CHUNKS; ++cc)
      s += part[((size_t)b * CHUNKS + cc) * TOK + c];
    actA[r * TOK + c] = s * (1.0f / (float)S_LEN);
  }
  __syncthreads();

  // Extractor: [16x60] @ [60x20] + b_ext (no relu) -> actB ld=32
  wmma_layer(actA, 64, EFFD, w_ext, b_ext, 20, actB, 32, false, lane, wave, 4);
  __syncthreads();
  // L0: 20 -> 256, relu -> actA ld=256
  wmma_layer(actB, 32, 20, w0, b0, 256, actA, 256, true, lane, wave, 4);
  __syncthreads();
  // L1: 256 -> 128, relu -> actB ld=128
  wmma_layer(actA, 256, 256, w1, b1, 128, actB, 128, true, lane, wave, 4);
  __syncthreads();
  // L2: 128 -> 64, relu -> actA ld=64
  wmma_layer(actB, 128, 128, w2, b2, 64, actA, 64, true, lane, wave, 4);
  __syncthreads();
  // L3: 64 -> 3 (no relu) -> actB ld=16 (cols 0..2 valid)
  wmma_layer(actA, 64, 64, w3, b3, 3, actB, 16, false, lane, wave, 4);
  __syncthreads();

  if (tid < 16 * 3) {
    const int r = tid / 3;
    const int c = tid % 3;
    out[(size_t)(mt * 16 + r) * 3 + c] = actB[r * 16 + c];
  }
}

// ---------------------------------------------------------------------------
extern "C" void kernel_launch(void* const* d_in, const int* in_sizes, int n_in,
                              void* d_out, int out_size, void* d_ws, size_t ws_size,
                              hipStream_t stream) {
  const float* x     = (const float*)d_in[0];
  const float* w_ext = (const float*)d_in[1];
  const float* b_ext = (const float*)d_in[2];
  const float* w0    = (const float*)d_in[3];
  const float* b0    = (const float*)d_in[4];
  const float* w1    = (const float*)d_in[5];
  const float* b1    = (const float*)d_in[6];
  const float* w2    = (const float*)d_in[7];
  const float* b2    = (const float*)d_in[8];
  const float* w3    = (const float*)d_in[9];
  const float* b3    = (const float*)d_in[10];
  float* out  = (float*)d_out;
  float* part = (float*)d_ws;   // B * CHUNKS * 64 floats = 512 KB

  const int B = in_sizes[0] / (S_LEN * TOK);   // 128

  dim3 g1(B, CHUNKS);
  mean_partial_kernel<<<g1, 256, 0, stream>>>(x, part);
  mlp_wmma_kernel<<<B / 16, 128, 0, stream>>>(part, w_ext, b_ext,
                                              w0, b0, w1, b1, w2, b2, w3, b3,
                                              out);
}